// DistortionAwareFeatureExtractor_23691039604949
// MI455X (gfx1250) — compile-verified
//
#include <hip/hip_runtime.h>

typedef __attribute__((ext_vector_type(2))) float v2f;
typedef __attribute__((ext_vector_type(4))) float v4f;
typedef __attribute__((ext_vector_type(8))) float v8f;

#define IC 4
#define H_DIM 256
#define W_DIM 256
#define TS 16
#define HALO 3
#define TW (TS + 2 * HALO)   // 22
#define NFEAT 16
#define PH 64
#define OD 64

__device__ __forceinline__ v8f wmma_f32_16x16x4(v2f a, v2f b, v8f c) {
  return __builtin_amdgcn_wmma_f32_16x16x4_f32(false, a, false, b, (short)0, c,
                                               false, false);
}

// ---------------------------------------------------------------------------
// Kernel 1: token projection  I[b,j,y,x] = sum_c tokens[n,c] * W_int[j,c]
// Memory-bound (188 MB token reads). Tokens are streamed exactly once ->
// non-temporal loads so they don't sweep the 192 MB L2 and evict I (4 MB),
// which kernel 2 re-reads. W_int indices are wave-uniform -> scalar loads.
// ---------------------------------------------------------------------------
__global__ __launch_bounds__(256) void proj_kernel(
    const float* __restrict__ tokens, const float* __restrict__ Wint,
    float* __restrict__ I, int N, int C) {
  int n = blockIdx.x * 256 + threadIdx.x;
  if (n >= N) return;
  const v4f* tp = (const v4f*)(tokens + (size_t)n * C);
  float acc0 = 0.f, acc1 = 0.f, acc2 = 0.f, acc3 = 0.f;
  const int C4 = C >> 2;  // C=180 -> 45
  for (int i = 0; i < C4; ++i) {
    v4f tv = __builtin_nontemporal_load(tp + i);  // TH_NT: stream, no L2 pollution
    const float* w0 = Wint + 0 * C + i * 4;
    const float* w1 = Wint + 1 * C + i * 4;
    const float* w2 = Wint + 2 * C + i * 4;
    const float* w3 = Wint + 3 * C + i * 4;
    acc0 += tv.x * w0[0] + tv.y * w0[1] + tv.z * w0[2] + tv.w * w0[3];
    acc1 += tv.x * w1[0] + tv.y * w1[1] + tv.z * w1[2] + tv.w * w1[3];
    acc2 += tv.x * w2[0] + tv.y * w2[1] + tv.z * w2[2] + tv.w * w2[3];
    acc3 += tv.x * w3[0] + tv.y * w3[1] + tv.z * w3[2] + tv.w * w3[3];
  }
  const int hw = H_DIM * W_DIM;
  int b = n / hw;
  int rem = n - b * hw;
  // Regular (RT) stores: I is 4 MB, we WANT it resident in L2 for kernel 2.
  I[(size_t)(b * IC + 0) * hw + rem] = acc0;
  I[(size_t)(b * IC + 1) * hw + rem] = acc1;
  I[(size_t)(b * IC + 2) * hw + rem] = acc2;
  I[(size_t)(b * IC + 3) * hw + rem] = acc3;
}

// ---------------------------------------------------------------------------
// Kernel 2: fused stencil stats -> 16 feats -> LayerNorm -> WMMA MLP -> out
// One block = 16x16 pixel tile (256 threads, 8 waves). Each wave runs the
// 16-pixel WMMA MLP (f32 16x16x4) for two tile rows.
// ---------------------------------------------------------------------------
__global__ __launch_bounds__(256) void stats_mlp_kernel(
    const float* __restrict__ I, const float* __restrict__ gamma,
    const float* __restrict__ beta, const float* __restrict__ W1,
    const float* __restrict__ b1, const float* __restrict__ W2,
    const float* __restrict__ b2, float* __restrict__ out) {
  __shared__ float tile[IC][TW][TW];        // 7744 B
  __shared__ float Zs[256][NFEAT + 1];      // 17408 B (pad 17: conflict-free)
  __shared__ float Hs[8][16][PH + 1];       // 33280 B (pad 65: conflict-free)

  const int tid = threadIdx.x;
  const int bz = blockIdx.z;
  const int x0 = blockIdx.x * TS, y0 = blockIdx.y * TS;

  // ---- load input tile with halo (zero padding => SAME conv semantics) ----
  for (int idx = tid; idx < IC * TW * TW; idx += 256) {
    int c = idx / (TW * TW);
    int r = (idx - c * TW * TW) / TW;
    int q = idx - c * TW * TW - r * TW;
    int gy = y0 + r - HALO, gx = x0 + q - HALO;
    float v = 0.f;
    if (gy >= 0 && gy < H_DIM && gx >= 0 && gx < W_DIM)
      v = I[((size_t)(bz * IC + c) * H_DIM + gy) * W_DIM + gx];
    tile[c][r][q] = v;
  }
  __syncthreads();

  // ---- per-pixel features + LayerNorm ----
  {
    const int ly = tid >> 4, lx = tid & 15;
    const int gy = y0 + ly, gx = x0 + lx;
    const int cy = ly + HALO, cx = lx + HALO;
    // count_include_pad=False window counts
    int cy3 = min(gy + 1, H_DIM - 1) - max(gy - 1, 0) + 1;
    int cx3 = min(gx + 1, W_DIM - 1) - max(gx - 1, 0) + 1;
    int cy7 = min(gy + 3, H_DIM - 1) - max(gy - 3, 0) + 1;
    int cx7 = min(gx + 3, W_DIM - 1) - max(gx - 3, 0) + 1;
    float inv3 = 1.f / (float)(cy3 * cx3);
    float inv7 = 1.f / (float)(cy7 * cx7);

    float fs[8] = {0, 0, 0, 0, 0, 0, 0, 0};
    float fq[8] = {0, 0, 0, 0, 0, 0, 0, 0};
#pragma unroll
    for (int c = 0; c < IC; ++c) {
#define TV(dy, dx) tile[c][cy + (dy)][cx + (dx)]
      float a00 = TV(-1, -1), a01 = TV(-1, 0), a02 = TV(-1, 1);
      float a10 = TV(0, -1), ctr = TV(0, 0), a12 = TV(0, 1);
      float a20 = TV(1, -1), a21 = TV(1, 0), a22 = TV(1, 1);
      float sgx = -a00 + a02 - 2.f * a10 + 2.f * a12 - a20 + a22;
      float sgy = -a00 - 2.f * a01 - a02 + a20 + 2.f * a21 + a22;
      float f[8];
      f[0] = sgx * sgx + sgy * sgy;  // grad energy
#pragma unroll
      for (int d = 1; d <= 3; ++d) {  // |laplacian|, dilation d
        float lap = TV(-d, 0) + TV(d, 0) + TV(0, -d) + TV(0, d) - 4.f * ctr;
        f[d] = fabsf(lap);
      }
      float s3 = 0.f, q3 = 0.f, s7 = 0.f, q7 = 0.f;
#pragma unroll
      for (int dy = -3; dy <= 3; ++dy)
#pragma unroll
        for (int dx = -3; dx <= 3; ++dx) {
          float v = TV(dy, dx);
          s7 += v;
          q7 += v * v;
          if (dy >= -1 && dy <= 1 && dx >= -1 && dx <= 1) {
            s3 += v;
            q3 += v * v;
          }
        }
#undef TV
      float m3 = s3 * inv3, m7 = s7 * inv7;
      f[4] = fabsf(ctr - m3);
      f[5] = fabsf(ctr - m7);
      f[6] = fmaxf(q3 * inv3 - m3 * m3, 0.f);
      f[7] = fmaxf(q7 * inv7 - m7 * m7, 0.f);
#pragma unroll
      for (int i = 0; i < 8; ++i) {
        fs[i] += f[i];
        fq[i] += f[i] * f[i];
      }
    }
    float z[16];
    float zsum = 0.f, zsq = 0.f;
#pragma unroll
    for (int i = 0; i < 8; ++i) {  // mean/std over IC=4 channels (biased var)
      float mu = fs[i] * 0.25f;
      float var = fmaxf(fq[i] * 0.25f - mu * mu, 0.f);
      float sd = sqrtf(var + 1e-6f);
      float z0 = log1pf(fmaxf(mu, 0.f));
      float z1 = log1pf(sd);
      z[2 * i] = z0;
      z[2 * i + 1] = z1;
      zsum += z0 + z1;
      zsq += z0 * z0 + z1 * z1;
    }
    float mu16 = zsum * (1.f / 16.f);
    float var16 = fmaxf(zsq * (1.f / 16.f) - mu16 * mu16, 0.f);
    float rs = rsqrtf(var16 + 1e-5f);
#pragma unroll
    for (int i = 0; i < 16; ++i)
      Zs[tid][i] = (z[i] - mu16) * rs * gamma[i] + beta[i];
  }
  __syncthreads();

  // ---- WMMA MLP: per wave, 16 pixels, h = GELU(Z@W1^T+b1), o = h@W2^T+b2 ---
  const int wave = tid >> 5;
  const int lane = tid & 31;
  const int lm = lane & 15;   // A: row M / B,C: col N
  const int lh = lane >> 4;   // half-wave selector
  const v8f zero8 = {0.f, 0.f, 0.f, 0.f, 0.f, 0.f, 0.f, 0.f};

  for (int g = 0; g < 2; ++g) {
    const int grp = wave + 8 * g;  // tile row handled by this wave
    // layer 1: M=16 pixels, K=16 feats, N=64
    v8f acc[4] = {zero8, zero8, zero8, zero8};
#pragma unroll
    for (int k = 0; k < 4; ++k) {
      const int kb = 4 * k + 2 * lh;
      v2f A;
      A.x = Zs[16 * grp + lm][kb];
      A.y = Zs[16 * grp + lm][kb + 1];
#pragma unroll
      for (int t = 0; t < 4; ++t) {
        const int n = 16 * t + lm;
        v2f Bv;
        Bv.x = W1[n * NFEAT + kb];      // B[k][n] = W1[n][k]
        Bv.y = W1[n * NFEAT + kb + 1];
        acc[t] = wmma_f32_16x16x4(A, Bv, acc[t]);
      }
    }
    // bias + exact GELU, restage into A-layout source (LDS)
#pragma unroll
    for (int t = 0; t < 4; ++t) {
      const int n = 16 * t + lm;
      const float bb = b1[n];
#pragma unroll
      for (int j = 0; j < 8; ++j) {
        float x = acc[t][j] + bb;
        Hs[wave][j + 8 * lh][n] =
            0.5f * x * (1.f + erff(x * 0.70710678118654752f));
      }
    }
    __syncthreads();
    // layer 2: M=16 pixels, K=64, N=64
    v8f acc2[4] = {zero8, zero8, zero8, zero8};
#pragma unroll
    for (int k = 0; k < 16; ++k) {
      const int kb = 4 * k + 2 * lh;
      v2f A;
      A.x = Hs[wave][lm][kb];
      A.y = Hs[wave][lm][kb + 1];
#pragma unroll
      for (int t = 0; t < 4; ++t) {
        const int n = 16 * t + lm;
        v2f Bv;
        Bv.x = W2[n * PH + kb];         // B[k][n] = W2[n][k]
        Bv.y = W2[n * PH + kb + 1];
        acc2[t] = wmma_f32_16x16x4(A, Bv, acc2[t]);
      }
    }
    // store: C reg j holds pixel x0+(j+8*lh) in tile row grp, feature 16t+lm
    // Output (67 MB) is write-only -> non-temporal stores, keep L2 for I.
    const int gy = y0 + grp;
#pragma unroll
    for (int t = 0; t < 4; ++t) {
      const int n = 16 * t + lm;
      const float bb = b2[n];
#pragma unroll
      for (int j = 0; j < 8; ++j) {
        const int gx = x0 + j + 8 * lh;
        size_t pix = (size_t)bz * (H_DIM * W_DIM) + (size_t)gy * W_DIM + gx;
        __builtin_nontemporal_store(acc2[t][j] + bb, &out[pix * OD + n]);
      }
    }
    __syncthreads();
  }
}

// ---------------------------------------------------------------------------
extern "C" void kernel_launch(void* const* d_in, const int* in_sizes, int n_in,
                              void* d_out, int out_size, void* d_ws,
                              size_t ws_size, hipStream_t stream) {
  const float* tokens = (const float*)d_in[0];
  const float* Wint = (const float*)d_in[1];
  const float* gamma = (const float*)d_in[2];
  const float* beta = (const float*)d_in[3];
  const float* W1 = (const float*)d_in[4];
  const float* b1 = (const float*)d_in[5];
  const float* W2 = (const float*)d_in[6];
  const float* b2 = (const float*)d_in[7];
  // d_in[8]/d_in[9] are H,W as device scalars; geometry fixed at 256 in setup.
  const int C = in_sizes[1] / IC;            // 180
  const int N = in_sizes[0] / C;             // B*H*W = 262144
  const int B = N / (H_DIM * W_DIM);         // 4

  float* I = (float*)d_ws;                   // [B, IC, H, W] = 4 MB, L2-resident

  proj_kernel<<<dim3((N + 255) / 256), dim3(256), 0, stream>>>(tokens, Wint, I,
                                                               N, C);
  dim3 grid(W_DIM / TS, H_DIM / TS, B);
  stats_mlp_kernel<<<grid, dim3(256), 0, stream>>>(I, gamma, beta, W1, b1, W2,
                                                   b2, (float*)d_out);
}